// AttnDecoderRNN_8194797600876
// MI455X (gfx1250) — compile-verified
//
#include <hip/hip_runtime.h>
#include <hip/hip_bf16.h>

// ---------------------------------------------------------------------------
// AttnDecoderRNN single step on gfx1250 (wave32, WMMA).
// All GEMVs are computed 16 rows per wave with V_WMMA_F32_16X16X4_F32:
//   A = 16x4 tile of W (per-lane global b64 load, streams 16 rows),
//   B = x[k..k+3] placed in column N=0 only (read from LDS via ds_load_b64),
//   D column 0 accumulates 16 dot products (lane 0: M=0..7, lane 16: M=8..15).
// K is a template parameter so the hot loop has no remainder handling.
// ---------------------------------------------------------------------------

typedef float v2f __attribute__((ext_vector_type(2)));
typedef float v8f __attribute__((ext_vector_type(8)));

#define HID    1024
#define VOCAB  50257
#define MAXLEN 2048

// ------------------------- wave-level reductions ---------------------------
__device__ __forceinline__ float wave_max(float v) {
  #pragma unroll
  for (int o = 16; o > 0; o >>= 1) v = fmaxf(v, __shfl_xor(v, o, 32));
  return v;
}
__device__ __forceinline__ float wave_sum(float v) {
  #pragma unroll
  for (int o = 16; o > 0; o >>= 1) v += __shfl_xor(v, o, 32);
  return v;
}

__device__ __forceinline__ float block_max(float v) {
  __shared__ float buf[32];
  __shared__ float res;
  const int lane = threadIdx.x & 31, w = threadIdx.x >> 5;
  const int nw = (blockDim.x + 31) >> 5;
  v = wave_max(v);
  if (lane == 0) buf[w] = v;
  __syncthreads();
  if (w == 0) {
    float t = (threadIdx.x < nw) ? buf[threadIdx.x] : -__builtin_inff();
    t = wave_max(t);
    if (threadIdx.x == 0) res = t;
  }
  __syncthreads();
  return res;
}

__device__ __forceinline__ float block_sum(float v) {
  __shared__ float buf[32];
  __shared__ float res;
  const int lane = threadIdx.x & 31, w = threadIdx.x >> 5;
  const int nw = (blockDim.x + 31) >> 5;
  v = wave_sum(v);
  if (lane == 0) buf[w] = v;
  __syncthreads();
  if (w == 0) {
    float t = (threadIdx.x < nw) ? buf[threadIdx.x] : 0.0f;
    t = wave_sum(t);
    if (threadIdx.x == 0) res = t;
  }
  __syncthreads();
  return res;
}

// --------------------- WMMA 16-rows-per-wave GEMV --------------------------
// y[row0..row0+15] = W[row0..row0+15, 0:K] @ xs + bias, optional relu.
// W row-major, leading dim == K. xs lives in LDS. EXEC full (uniform flow).
template <int K>
__device__ __forceinline__ void wave_gemv16(const float* __restrict__ W,
                                            long long row0, long long nrows,
                                            const float* __restrict__ xs,
                                            const float* __restrict__ bias,
                                            float* __restrict__ y, bool relu) {
  const int lane = threadIdx.x & 31;
  const int m    = lane & 15;        // A-matrix row within the 16-row tile
  const int half = lane >> 4;        // 0 -> K pair {0,1}, 1 -> K pair {2,3}
  long long r = row0 + m;
  if (r > nrows - 1) r = nrows - 1;  // clamp tail rows to stay in-bounds
  const float* wp = W + r * (long long)K + 2 * half;
  const float* xp = xs + 2 * half;
  const bool n0 = (m == 0);          // lanes 0 and 16 carry B column N=0

  v8f acc = {};
  #pragma unroll 8
  for (int k = 0; k < K; k += 4) {
    v2f a  = *(const v2f*)(wp + k);                // A[m][k+2h], A[m][k+2h+1]
    v2f xv = *(const v2f*)(xp + k);                // xs[k+2h], xs[k+2h+1] (LDS)
    v2f b;
    b.x = n0 ? xv.x : 0.0f;                        // B[k'][0] only in col 0
    b.y = n0 ? xv.y : 0.0f;
    acc = __builtin_amdgcn_wmma_f32_16x16x4_f32(
        /*neg_a=*/false, a, /*neg_b=*/false, b,
        /*c_mod=*/(short)0, acc, /*reuse_a=*/false, /*reuse_b=*/false);
  }

  if (n0) {                                        // lane 0: M=0..7, lane16: M=8..15
    const int mbase = half * 8;
    #pragma unroll
    for (int i = 0; i < 8; ++i) {
      long long rr = row0 + mbase + i;
      if (rr < nrows) {
        float v = acc[i] + bias[rr];
        if (relu) v = v > 0.0f ? v : 0.0f;
        y[rr] = v;
      }
    }
  }
}

template <int K>
__global__ void gemv_wmma_kernel(const float* __restrict__ W, long long nrows,
                                 const float* __restrict__ x,
                                 const float* __restrict__ bias,
                                 float* __restrict__ y, int relu) {
  __shared__ float xs[K];                          // shared B-vector, once per block
  for (int i = threadIdx.x; i < K; i += blockDim.x) xs[i] = x[i];
  __syncthreads();

  const long long wavesPerBlock = blockDim.x >> 5;
  const long long g = (long long)blockIdx.x * wavesPerBlock + (threadIdx.x >> 5);
  const long long row0 = g * 16;
  if (row0 < nrows)                                // wave-uniform guard
    wave_gemv16<K>(W, row0, nrows, xs, bias, y, relu != 0);
}

// Both GRU weight matrices in one launch: rows [0,3072)->W_ih@x, [3072,6144)->W_hh@h
__global__ void gru_gates_kernel(const float* __restrict__ W_ih,
                                 const float* __restrict__ W_hh,
                                 const float* __restrict__ x,
                                 const float* __restrict__ h,
                                 const float* __restrict__ b_ih,
                                 const float* __restrict__ b_hh,
                                 float* __restrict__ gi, float* __restrict__ gh) {
  __shared__ float xs[2 * HID];                    // [0,HID)=x, [HID,2HID)=h
  for (int i = threadIdx.x; i < HID; i += blockDim.x) {
    xs[i] = x[i];
    xs[HID + i] = h[i];
  }
  __syncthreads();

  const long long wavesPerBlock = blockDim.x >> 5;
  const long long g = (long long)blockIdx.x * wavesPerBlock + (threadIdx.x >> 5);
  const long long row0 = g * 16;
  if (row0 < 3 * HID) {
    wave_gemv16<HID>(W_ih, row0, 3 * HID, xs, b_ih, gi, false);
  } else if (row0 < 6 * HID) {
    wave_gemv16<HID>(W_hh, row0 - 3 * HID, 3 * HID, xs + HID, b_hh, gh, false);
  }
}

// --------------------------- elementwise kernels ---------------------------
__global__ void build_inputs_kernel(const int* __restrict__ tok,
                                    const float* __restrict__ hidden,
                                    const float* __restrict__ emb,
                                    float* __restrict__ xh,   // [2048] = emb ++ h
                                    float* __restrict__ x2) { // [2048] = emb ++ attn_applied
  const int i = blockIdx.x * blockDim.x + threadIdx.x;
  if (i < HID) {
    float e = emb[(long long)tok[0] * HID + i];
    xh[i] = e;
    x2[i] = e;
  } else if (i < 2 * HID) {
    xh[i] = hidden[i - HID];
  }
}

__global__ void attn_softmax_kernel(const float* __restrict__ logits,
                                    float* __restrict__ weights) { // 1 block x 256
  const int tid = threadIdx.x;
  float m = -__builtin_inff();
  for (int i = tid; i < MAXLEN; i += 256) m = fmaxf(m, logits[i]);
  m = block_max(m);
  float s = 0.0f;
  for (int i = tid; i < MAXLEN; i += 256) s += __expf(logits[i] - m);
  s = block_sum(s);
  const float inv = 1.0f / s;
  for (int i = tid; i < MAXLEN; i += 256) weights[i] = __expf(logits[i] - m) * inv;
}

// attn_applied = weights @ encoder_outputs, split-K over 16 blocks of 128 rows.
__global__ void attn_apply_partial_kernel(const float* __restrict__ wts,
                                          const float* __restrict__ enc,
                                          float* __restrict__ partials) { // [16][1024]
  const int b = blockIdx.x, tid = threadIdx.x;
  float acc0 = 0.f, acc1 = 0.f, acc2 = 0.f, acc3 = 0.f;
  for (int j = 0; j < 128; ++j) {
    const int row = b * 128 + j;
    const float wj = wts[row];
    const float* er = enc + (long long)row * HID;
    acc0 += wj * er[tid];
    acc1 += wj * er[tid + 256];
    acc2 += wj * er[tid + 512];
    acc3 += wj * er[tid + 768];
  }
  float* p = partials + (long long)b * HID;
  p[tid] = acc0; p[tid + 256] = acc1; p[tid + 512] = acc2; p[tid + 768] = acc3;
}

__global__ void attn_apply_reduce_kernel(const float* __restrict__ partials,
                                         float* __restrict__ x2) {
  const int c = blockIdx.x * blockDim.x + threadIdx.x;
  if (c < HID) {
    float s = 0.0f;
    #pragma unroll
    for (int b = 0; b < 16; ++b) s += partials[b * HID + c];
    x2[HID + c] = s;  // x2 = [embedded, attn_applied]
  }
}

__global__ void gru_combine_kernel(const float* __restrict__ gi,
                                   const float* __restrict__ gh,
                                   const float* __restrict__ h,
                                   float* __restrict__ hnew) {
  const int i = blockIdx.x * blockDim.x + threadIdx.x;
  if (i < HID) {
    const float r = 1.0f / (1.0f + __expf(-(gi[i] + gh[i])));
    const float z = 1.0f / (1.0f + __expf(-(gi[HID + i] + gh[HID + i])));
    const float n = tanhf(gi[2 * HID + i] + r * gh[2 * HID + i]);
    hnew[i] = (1.0f - z) * n + z * h[i];
  }
}

// ------------------------------ log-softmax --------------------------------
__global__ void lse_partial_max_kernel(const float* __restrict__ l, int n,
                                       float* __restrict__ pmax) {
  float m = -__builtin_inff();
  for (int i = blockIdx.x * blockDim.x + threadIdx.x; i < n;
       i += gridDim.x * blockDim.x)
    m = fmaxf(m, l[i]);
  m = block_max(m);
  if (threadIdx.x == 0) pmax[blockIdx.x] = m;
}

__global__ void lse_finish_kernel(const float* __restrict__ l, int n,
                                  const float* __restrict__ pmax, int npmax,
                                  float* __restrict__ stats) { // 1 block x 1024
  __shared__ float sm;
  float m = -__builtin_inff();
  for (int i = threadIdx.x; i < npmax; i += blockDim.x) m = fmaxf(m, pmax[i]);
  m = block_max(m);
  if (threadIdx.x == 0) sm = m;
  __syncthreads();
  m = sm;
  float s = 0.0f;
  for (int i = threadIdx.x; i < n; i += blockDim.x) s += __expf(l[i] - m);
  s = block_sum(s);
  if (threadIdx.x == 0) { stats[0] = m; stats[1] = logf(s); }
}

__global__ void lse_apply_kernel(const float* __restrict__ l, int n,
                                 const float* __restrict__ stats,
                                 float* __restrict__ out) {
  const int i = blockIdx.x * blockDim.x + threadIdx.x;
  if (i < n) out[i] = l[i] - stats[0] - stats[1];
}

// ------------------------------- launcher ----------------------------------
extern "C" void kernel_launch(void* const* d_in, const int* in_sizes, int n_in,
                              void* d_out, int out_size, void* d_ws, size_t ws_size,
                              hipStream_t stream) {
  const int*   tok    = (const int*)  d_in[0];
  const float* hidden = (const float*)d_in[1];
  const float* enc    = (const float*)d_in[2];
  const float* emb    = (const float*)d_in[3];
  const float* attn_W = (const float*)d_in[4];
  const float* attn_b = (const float*)d_in[5];
  const float* comb_W = (const float*)d_in[6];
  const float* comb_b = (const float*)d_in[7];
  const float* W_ih   = (const float*)d_in[8];
  const float* W_hh   = (const float*)d_in[9];
  const float* b_ih   = (const float*)d_in[10];
  const float* b_hh   = (const float*)d_in[11];
  const float* out_W  = (const float*)d_in[12];
  const float* out_b  = (const float*)d_in[13];

  float* out = (float*)d_out;            // [0,50257) log-probs | [50257,51281) h_new | [51281,53329) attn_w
  float* ws  = (float*)d_ws;

  float* xh         = ws;                // 2048
  float* x2         = ws + 2048;         // 2048
  float* attnLogits = ws + 4096;         // 2048
  float* partials   = ws + 6144;         // 16*1024
  float* xrelu      = ws + 22528;        // 1024
  float* gi         = ws + 23552;        // 3072
  float* gh         = ws + 26624;        // 3072
  float* bigLogits  = ws + 29696;        // 50257
  float* pmax       = ws + 79960;        // 128
  float* stats      = ws + 80092;        // 2

  float* hnew_out   = out + VOCAB;       // 1024
  float* attnw_out  = out + VOCAB + HID; // 2048

  // 1) embedded + concat vectors
  build_inputs_kernel<<<8, 256, 0, stream>>>(tok, hidden, emb, xh, x2);
  // 2) attention logits: 2048x2048 GEMV via WMMA (128 waves)
  gemv_wmma_kernel<2 * HID><<<16, 256, 0, stream>>>(attn_W, MAXLEN, xh, attn_b,
                                                    attnLogits, 0);
  // 3) softmax over 2048 -> attn_weights output
  attn_softmax_kernel<<<1, 256, 0, stream>>>(attnLogits, attnw_out);
  // 4) attn_applied = weights @ enc (split-K partials, deterministic reduce)
  attn_apply_partial_kernel<<<16, 256, 0, stream>>>(attnw_out, enc, partials);
  attn_apply_reduce_kernel<<<4, 256, 0, stream>>>(partials, x2);
  // 5) combine + relu: 1024x2048 GEMV via WMMA (64 waves)
  gemv_wmma_kernel<2 * HID><<<8, 256, 0, stream>>>(comb_W, HID, x2, comb_b,
                                                   xrelu, 1);
  // 6) GRU gates: two 3072x1024 GEMVs via WMMA in one launch (384 waves)
  gru_gates_kernel<<<48, 256, 0, stream>>>(W_ih, W_hh, xrelu, hidden, b_ih, b_hh,
                                           gi, gh);
  // 7) gate nonlinearity -> h_new output
  gru_combine_kernel<<<4, 256, 0, stream>>>(gi, gh, hidden, hnew_out);
  // 8) vocab projection: 50257x1024 GEMV via WMMA (3142 waves)
  gemv_wmma_kernel<HID><<<393, 256, 0, stream>>>(out_W, VOCAB, hnew_out, out_b,
                                                 bigLogits, 0);
  // 9) log-softmax over 50257
  lse_partial_max_kernel<<<128, 256, 0, stream>>>(bigLogits, VOCAB, pmax);
  lse_finish_kernel<<<1, 1024, 0, stream>>>(bigLogits, VOCAB, pmax, 128, stats);
  lse_apply_kernel<<<197, 256, 0, stream>>>(bigLogits, VOCAB, stats, out);
}